// GNN_91061896609816
// MI455X (gfx1250) — compile-verified
//
#include <hip/hip_runtime.h>
#include <hip/hip_bf16.h>

typedef __attribute__((ext_vector_type(2))) float v2f;
typedef __attribute__((ext_vector_type(8))) float v8f;

#define HD 16          // hidden dim (both GCN layers)
#define NCLS 7         // output classes
#define MAXK 256       // max reduction dim (NUM_FEATURES)

// ---------------- degree / normalization ----------------

__global__ void k_init_deg(float* __restrict__ deg, int n) {
  int i = blockIdx.x * blockDim.x + threadIdx.x;
  if (i < n) deg[i] = 1.0f;                       // self-loop contributes 1
}

__global__ void k_deg_count(const int* __restrict__ dst, float* __restrict__ deg, int e) {
  int i = blockIdx.x * blockDim.x + threadIdx.x;
  if (i < e) unsafeAtomicAdd(&deg[dst[i]], 1.0f); // native global_atomic_add_f32
}

__global__ void k_deg_rsqrt(float* __restrict__ deg, int n) {
  int i = blockIdx.x * blockDim.x + threadIdx.x;
  if (i < n) deg[i] = rsqrtf(deg[i]);             // deg >= 1 always
}

// ---------------- dense GEMM: Y[n x 16] = X[n x K] @ W[K x 16] via WMMA f32 ----------------
// One wave per 16-row tile. A 16x4 layout: lanes 0-15 hold K={0,1}, lanes 16-31 K={2,3},
// M = lane&15. B 4x16: same K split, N = lane&15. C/D: VGPR v holds row v + 8*(lane>>4).
// W staged in LDS (ds_load_2addr_b32, conflict-free). #pragma unroll 4 keeps 4 A-loads
// in flight per wave to hide the strided-gather latency.

__global__ void k_gemm_wmma(const float* __restrict__ X, const float* __restrict__ W,
                            float* __restrict__ Y, int n_rows, int K) {
  __shared__ float Wlds[MAXK * HD];

  // cooperative stage of W (all threads participate BEFORE any wave exits)
  for (int i = threadIdx.x; i < K * HD; i += blockDim.x) Wlds[i] = W[i];
  __syncthreads();

  int wave = (blockIdx.x * blockDim.x + threadIdx.x) >> 5;
  int lane = threadIdx.x & 31;
  int row0 = wave * 16;
  if (row0 >= n_rows) return;                     // uniform per-wave: EXEC all-ones for WMMA

  const int m    = row0 + (lane & 15);
  const int kb   = (lane >> 4) << 1;              // 0 or 2
  const int ncol = lane & 15;
  const float* Arow = X + (size_t)m * K + kb;     // 8-byte aligned (kb even, k0 % 4 == 0)

  v8f c = {};
  const int kmain = (K > 64) ? (K - 64) : 0;      // 48 iters for K=256 (divisible by 4)
  int k0 = 0;
#pragma unroll 4
  for (; k0 < kmain; k0 += 4) {                   // main body: unconditional prefetch
    __builtin_prefetch(Arow + k0 + 64, 0, 3);     // global_prefetch_b8
    v2f a = *(const v2f*)(Arow + k0);
    v2f b;
    b.x = Wlds[(k0 + kb)     * HD + ncol];
    b.y = Wlds[(k0 + kb + 1) * HD + ncol];
    c = __builtin_amdgcn_wmma_f32_16x16x4_f32(false, a, false, b, (short)0, c,
                                              false, false);
  }
#pragma unroll 4
  for (; k0 < K; k0 += 4) {                       // tail (last 64 k's; all of K if K <= 64): fully pipelined
    v2f a = *(const v2f*)(Arow + k0);
    v2f b;
    b.x = Wlds[(k0 + kb)     * HD + ncol];
    b.y = Wlds[(k0 + kb + 1) * HD + ncol];
    c = __builtin_amdgcn_wmma_f32_16x16x4_f32(false, a, false, b, (short)0, c,
                                              false, false);
  }

  const int half = lane >> 4;
#pragma unroll
  for (int v = 0; v < 8; ++v)
    Y[(size_t)(row0 + v + 8 * half) * HD + ncol] = c[v];        // 64B coalesced per half-wave
}

// ---------------- aggregation ----------------

// agg[i][:] = xw[i][:] * dinv[i]^2  (self-loop term; also initializes the buffer)
// one thread per 4 features
__global__ void k_init_agg(const float* __restrict__ xw, const float* __restrict__ dinv,
                           float* __restrict__ agg, int n) {
  int t = blockIdx.x * blockDim.x + threadIdx.x;  // n * (HD/4) threads
  if (t >= n * (HD / 4)) return;
  float di = dinv[t >> 2];
  float s  = di * di;
  float4 v = ((const float4*)xw)[t];
  v.x *= s; v.y *= s; v.z *= s; v.w *= s;
  ((float4*)agg)[t] = v;
}

// 4 lanes per edge, 4 features each: agg[dst][j] += xw[src][j] * dinv[src]*dinv[dst]
__global__ void k_edge_agg(const int* __restrict__ src, const int* __restrict__ dst,
                           const float* __restrict__ dinv, const float* __restrict__ xw,
                           float* __restrict__ agg, int e) {
  int t = blockIdx.x * blockDim.x + threadIdx.x;
  int eid = t >> 2;
  if (eid >= e) return;
  int q = (t & 3) << 2;                            // feature base: 0,4,8,12
  int s = src[eid];
  int d = dst[eid];
  float nrm = dinv[s] * dinv[d];
  float4 v = *(const float4*)(xw + (size_t)s * HD + q);   // 64B coalesced per edge
  float* ap = agg + (size_t)d * HD + q;
  unsafeAtomicAdd(ap + 0, v.x * nrm);
  unsafeAtomicAdd(ap + 1, v.y * nrm);
  unsafeAtomicAdd(ap + 2, v.z * nrm);
  unsafeAtomicAdd(ap + 3, v.w * nrm);
}

// h = relu(agg + b), in place; one thread per 4 features
__global__ void k_bias_relu(float* __restrict__ agg, const float* __restrict__ b, int n) {
  int t = blockIdx.x * blockDim.x + threadIdx.x;
  if (t >= n * (HD / 4)) return;
  int q = (t & 3) << 2;
  float4 v = ((const float4*)agg)[t];
  v.x = fmaxf(v.x + b[q + 0], 0.0f);
  v.y = fmaxf(v.y + b[q + 1], 0.0f);
  v.z = fmaxf(v.z + b[q + 2], 0.0f);
  v.w = fmaxf(v.w + b[q + 3], 0.0f);
  ((float4*)agg)[t] = v;
}

// ---------------- output head: logits = h @ Wout + bout ; log_softmax ----------------

__global__ void k_out(const float* __restrict__ H, const float* __restrict__ Wout,
                      const float* __restrict__ bout, float* __restrict__ out, int n) {
  int i = blockIdx.x * blockDim.x + threadIdx.x;
  if (i >= n) return;
  float h[HD];
#pragma unroll
  for (int k = 0; k < HD; ++k) h[k] = H[(size_t)i * HD + k];
  float l[NCLS];
#pragma unroll
  for (int c = 0; c < NCLS; ++c) {
    float acc = bout[c];
#pragma unroll
    for (int k = 0; k < HD; ++k) acc = fmaf(h[k], Wout[k * NCLS + c], acc);
    l[c] = acc;
  }
  float m = l[0];
#pragma unroll
  for (int c = 1; c < NCLS; ++c) m = fmaxf(m, l[c]);
  float s = 0.0f;
#pragma unroll
  for (int c = 0; c < NCLS; ++c) s += expf(l[c] - m);
  float lse = m + logf(s);
#pragma unroll
  for (int c = 0; c < NCLS; ++c) out[(size_t)i * NCLS + c] = l[c] - lse;
}

// ---------------- launch ----------------

extern "C" void kernel_launch(void* const* d_in, const int* in_sizes, int n_in,
                              void* d_out, int out_size, void* d_ws, size_t ws_size,
                              hipStream_t stream) {
  const float* x    = (const float*)d_in[0];
  const int*   ei   = (const int*)  d_in[1];   // [2, E] row-major
  const float* W1   = (const float*)d_in[2];
  const float* b1   = (const float*)d_in[3];
  const float* W2   = (const float*)d_in[4];
  const float* b2   = (const float*)d_in[5];
  const float* Wout = (const float*)d_in[6];
  const float* bout = (const float*)d_in[7];

  const int F = in_sizes[2] / HD;      // 256 features
  const int N = in_sizes[0] / F;       // 100000 nodes
  const int E = in_sizes[1] / 2;       // 3200000 edges

  const int* src = ei;
  const int* dst = ei + E;

  // workspace layout (floats): deg | bufA | bufB | bufC   (~20 MB total)
  float* ws   = (float*)d_ws;
  float* deg  = ws;                                    // N (becomes dinv)
  float* bufA = ws + ((size_t)(N + 255) / 256) * 256;  // N*HD  (xw of current layer)
  float* bufB = bufA + (size_t)N * HD;                 // N*HD  (agg1 -> h1)
  float* bufC = bufB + (size_t)N * HD;                 // N*HD  (agg2 -> h2)

  dim3 blk(256);
  int nb_n  = (N + 255) / 256;
  int nb_e  = (E + 255) / 256;
  int nb_n4 = (N * (HD / 4) + 255) / 256;                       // float4 streaming kernels
  int nb_e4 = (int)(((long long)E * 4 + 255) / 256);            // 4 lanes per edge
  int tiles = (N + 15) / 16;
  int nb_g  = (tiles + 7) / 8;   // 8 waves (256 threads) per block

  // normalization
  k_init_deg <<<nb_n, blk, 0, stream>>>(deg, N);
  k_deg_count<<<nb_e, blk, 0, stream>>>(dst, deg, E);
  k_deg_rsqrt<<<nb_n, blk, 0, stream>>>(deg, N);

  // layer 1
  k_gemm_wmma<<<nb_g,  blk, 0, stream>>>(x, W1, bufA, N, F);
  k_init_agg <<<nb_n4, blk, 0, stream>>>(bufA, deg, bufB, N);
  k_edge_agg <<<nb_e4, blk, 0, stream>>>(src, dst, deg, bufA, bufB, E);
  k_bias_relu<<<nb_n4, blk, 0, stream>>>(bufB, b1, N);

  // layer 2
  k_gemm_wmma<<<nb_g,  blk, 0, stream>>>(bufB, W2, bufA, N, HD);
  k_init_agg <<<nb_n4, blk, 0, stream>>>(bufA, deg, bufC, N);
  k_edge_agg <<<nb_e4, blk, 0, stream>>>(src, dst, deg, bufA, bufC, E);
  k_bias_relu<<<nb_n4, blk, 0, stream>>>(bufC, b2, N);

  // head
  k_out<<<nb_n, blk, 0, stream>>>(bufC, Wout, bout, (float*)d_out, N);
}